// PostprocessExport_53618371723810
// MI455X (gfx1250) — compile-verified
//
#include <hip/hip_runtime.h>
#include <stdint.h>

constexpr int   B          = 32;
constexpr int   N          = 25200;
constexpr int   CH         = 85;
constexpr int   NCLS       = 80;
constexpr int   TOPK       = 512;
constexpr float CONF_THRE  = 0.7f;
constexpr float NMS_THRE   = 0.45f;
constexpr float CLASS_OFF  = 4096.0f;
constexpr uint32_t KEY_THRE = 0x3F333333u;  // float bits of 0.7f
constexpr int   SORTN      = 2048;          // bitonic sort width in k_select

typedef float v2f __attribute__((ext_vector_type(2)));
typedef float v8f __attribute__((ext_vector_type(8)));

// --------------------------------------------------------------------------
// Pass 1 (bandwidth-bound, 274 MB): wave-per-anchor coalesced read of the 85
// channels; wave max-reduction over the 80 class scores; one conf key/anchor.
// --------------------------------------------------------------------------
__global__ void k_scores(const float* __restrict__ pred, uint32_t* __restrict__ keys) {
  int gid  = blockIdx.x * blockDim.x + threadIdx.x;
  int wave = gid >> 5;
  int lane = gid & 31;
  if (wave >= B * N) return;
  const float* p = pred + (size_t)wave * CH;
  float v0 = p[lane];                               // c = 0..31
  float v1 = p[32 + lane];                          // c = 32..63
  float v2 = (64 + lane < CH) ? p[64 + lane] : 0.f; // c = 64..84
  float m  = (lane >= 5) ? v0 : -1.0f;              // classes start at c=5
  m = fmaxf(m, v1);
  if (64 + lane < CH) m = fmaxf(m, v2);
#pragma unroll
  for (int off = 16; off > 0; off >>= 1) m = fmaxf(m, __shfl_xor(m, off, 32));
  float obj = __shfl(v0, 4, 32);
  if (lane == 0) keys[wave] = __float_as_uint(obj * m);
}

// --------------------------------------------------------------------------
// Top-512 select per image. All valid keys are in [0.7,1.0) => exponent 126,
// so (key>>11)&0xFFF is a monotone 12-bit mantissa bin. Histogram radix-select
// the boundary bin, collect sure+tie candidates, bitonic-sort 64-bit
// composites (key<<32 | (0xFFFFFFFF-idx)) descending -> exact top_k order
// with smallest-index tie-break.
// --------------------------------------------------------------------------
__device__ __forceinline__ int bin_of(uint32_t kv) {
  return (kv >= 0x3F800000u) ? 0xFFF : (int)((kv >> 11) & 0xFFFu);
}

__global__ void k_select(const uint32_t* __restrict__ keys, int* __restrict__ selIdx) {
  const int img = blockIdx.x;
  const int tid = threadIdx.x;
  const int bs  = blockDim.x;
  __shared__ uint32_t hist[4096];
  __shared__ unsigned long long sbuf[SORTN];
  __shared__ int nSure, nTie, Tbin;

  for (int i = tid; i < 4096; i += bs) hist[i] = 0u;
  if (tid == 0) { nSure = 0; nTie = 0; }
  __syncthreads();

  const uint32_t* k = keys + (size_t)img * N;
  for (int i = tid; i < N; i += bs) {
    uint32_t kv = k[i];
    if (kv >= KEY_THRE) atomicAdd(&hist[bin_of(kv)], 1u);
  }
  __syncthreads();

  if (tid == 0) {
    int cum = 0, T = -1;
    for (int b = 4095; b >= 0; --b) {
      cum += (int)hist[b];
      if (cum >= TOPK) { T = b; break; }
    }
    Tbin = T;
  }
  __syncthreads();
  const int T = Tbin;

  for (int i = tid; i < SORTN; i += bs) sbuf[i] = ~0ull;
  __syncthreads();

  for (int i = tid; i < N; i += bs) {
    uint32_t kv = k[i];
    if (kv < KEY_THRE) continue;
    int b = bin_of(kv);
    unsigned long long comp =
        ~((((unsigned long long)kv) << 32) | (unsigned long long)(0xFFFFFFFFu - (uint32_t)i));
    if (b > T) {
      int p = atomicAdd(&nSure, 1);          // guaranteed < 512 by T construction
      sbuf[p] = comp;
    } else if (b == T) {
      int p = atomicAdd(&nTie, 1);
      if (p < SORTN - TOPK) sbuf[TOPK + p] = comp;
    }
  }
  __syncthreads();

  // bitonic sort ascending on complemented composites == descending on (key,~idx)
  for (int kk = 2; kk <= SORTN; kk <<= 1) {
    for (int j = kk >> 1; j > 0; j >>= 1) {
      for (int i = tid; i < SORTN; i += bs) {
        int ixj = i ^ j;
        if (ixj > i) {
          unsigned long long a = sbuf[i], c = sbuf[ixj];
          bool up = ((i & kk) == 0);
          if (up ? (a > c) : (a < c)) { sbuf[i] = c; sbuf[ixj] = a; }
        }
      }
      __syncthreads();
    }
  }

  for (int s = tid; s < TOPK; s += bs) {
    unsigned long long comp = sbuf[s];
    if (comp == ~0ull) { selIdx[(size_t)img * TOPK + s] = -1; }
    else {
      unsigned long long e = ~comp;
      uint32_t idx = 0xFFFFFFFFu - (uint32_t)(e & 0xFFFFFFFFull);
      selIdx[(size_t)img * TOPK + s] = (int)idx;
    }
  }
}

// --------------------------------------------------------------------------
// Gather the 512 winners: boxes, class argmax (first-max), conf, offset NMS
// boxes, areas; wave32 ballot builds the initial keep bitmask.
// --------------------------------------------------------------------------
__global__ void k_gather(const float* __restrict__ pred, const int* __restrict__ selIdx,
                         float4* __restrict__ boxes, float4* __restrict__ nmsB,
                         float* __restrict__ areas, float* __restrict__ confs,
                         float* __restrict__ clsf, uint32_t* __restrict__ keepWords) {
  const int img = blockIdx.x;
  const int t   = threadIdx.x;  // 512 threads
  int idx = selIdx[(size_t)img * TOPK + t];
  float4 bb = make_float4(0.f, 0.f, 0.f, 0.f);
  float4 nb = bb;
  float area = 0.f, conf = 0.f, cls = 0.f;
  bool valid = false;
  if (idx >= 0) {
    const float* p = pred + ((size_t)img * N + (size_t)idx) * CH;
    float cx = p[0], cy = p[1], w = p[2], h = p[3], obj = p[4];
    float cmax = p[5];
    int ci = 0;
    for (int c = 1; c < NCLS; ++c) {
      float v = p[5 + c];
      if (v > cmax) { cmax = v; ci = c; }
    }
    conf  = obj * cmax;
    valid = (conf >= CONF_THRE);
    float hw = 0.5f * w, hh = 0.5f * h;
    bb = make_float4(cx - hw, cy - hh, cx + hw, cy + hh);
    float off = (float)ci * CLASS_OFF;
    nb = make_float4(bb.x + off, bb.y + off, bb.z + off, bb.w + off);
    area = (bb.z - bb.x) * (bb.w - bb.y);
    cls  = (float)ci;
  }
  size_t o = (size_t)img * TOPK + t;
  boxes[o] = bb; nmsB[o] = nb; areas[o] = area; confs[o] = conf; clsf[o] = cls;
  unsigned long long bal = __ballot(valid);
  if ((t & 31) == 0) keepWords[img * 16 + (t >> 5)] = (uint32_t)bal;
}

// --------------------------------------------------------------------------
// 512x512 IoU > thre bitmask, in 16x16 WMMA tiles. The union term
// area_i + area_j is a rank-2 outer sum computed on the matrix pipe with
// V_WMMA_F32_16X16X4_F32; VALU does the min/max intersection; a ballot per
// C-VGPR converts predicates straight into the bitmask row layout.
// Block = 256 threads = 8 waves; block handles one 16-row strip (tile row ti),
// each wave covers 4 column tiles.
// --------------------------------------------------------------------------
__global__ void k_iou(const float4* __restrict__ nmsB, const float* __restrict__ areas,
                      unsigned short* __restrict__ mask) {
  const int blk = blockIdx.x;
  const int img = blk >> 5;
  const int ti  = blk & 31;
  const int tid = threadIdx.x;
  __shared__ float4 colB[TOPK];
  __shared__ float  colA[TOPK];
  __shared__ float4 rowB[16];
  __shared__ float  rowA[16];

  const float4* nb = nmsB + (size_t)img * TOPK;
  const float*  ar = areas + (size_t)img * TOPK;
  for (int i = tid; i < TOPK; i += 256) { colB[i] = nb[i]; colA[i] = ar[i]; }
  if (tid < 16) { rowB[tid] = nb[ti * 16 + tid]; rowA[tid] = ar[ti * 16 + tid]; }
  __syncthreads();

  const int wave = tid >> 5;
  const int lane = tid & 31;
  const int l15  = lane & 15;
  const bool lo  = (lane < 16);

  // A: 16x4, col0 = row areas, col1 = ones (K=2,3 zero -> lanes 16-31 zero)
  v2f a;
  a.x = lo ? rowA[l15] : 0.0f;
  a.y = lo ? 1.0f      : 0.0f;

#pragma unroll
  for (int kk = 0; kk < 4; ++kk) {
    const int tj = wave * 4 + kk;
    // B: 4x16, row0 = ones, row1 = col areas
    v2f b;
    b.x = lo ? 1.0f               : 0.0f;
    b.y = lo ? colA[tj * 16 + l15] : 0.0f;
    v8f c = {};
    c = __builtin_amdgcn_wmma_f32_16x16x4_f32(false, a, false, b, (short)0, c, false, false);

    const int colIdx = tj * 16 + l15;
    const float4 cb  = colB[colIdx];
    const int rbase  = lo ? 0 : 8;
    const int i0     = ti * 16;
#pragma unroll
    for (int v = 0; v < 8; ++v) {
      float4 rb = rowB[rbase + v];
      float iw = fminf(rb.z, cb.z) - fmaxf(rb.x, cb.x); iw = fmaxf(iw, 0.0f);
      float ih = fminf(rb.w, cb.w) - fmaxf(rb.y, cb.y); ih = fmaxf(ih, 0.0f);
      float inter = iw * ih;
      float uni   = c[v] - inter;                    // area_i + area_j - inter (WMMA)
      bool p = inter > NMS_THRE * uni;
      unsigned long long bal = __ballot(p);
      // ballot bits[15:0] = row v (lanes 0-15), bits[31:16] = row v+8
      if (lane == v)
        mask[((size_t)img * TOPK + (i0 + v)) * 32 + tj]     = (unsigned short)(bal & 0xFFFFull);
      if (lane == v + 8)
        mask[((size_t)img * TOPK + (i0 + 8 + v)) * 32 + tj] = (unsigned short)((bal >> 16) & 0xFFFFull);
    }
  }
}

// --------------------------------------------------------------------------
// Greedy NMS (512 sequential steps) over LDS-resident bitmask; 16 keep-words
// live in wave-0 registers. Then 512 threads emit det (x6) and keep.
// --------------------------------------------------------------------------
__global__ void k_nms(const unsigned short* __restrict__ mask,
                      const uint32_t* __restrict__ keepWords,
                      const float4* __restrict__ boxes, const float* __restrict__ confs,
                      const float* __restrict__ clsf,
                      float* __restrict__ outDet, float* __restrict__ outKeep) {
  const int img = blockIdx.x;
  const int t   = threadIdx.x;  // 512 threads
  __shared__ uint32_t rowsL[TOPK * 16];  // 32 KB suppression bitmask
  __shared__ uint32_t keepL[16];

  const uint32_t* m32 = (const uint32_t*)mask + (size_t)img * TOPK * 16;
  for (int i = t; i < TOPK * 16; i += 512) rowsL[i] = m32[i];
  __syncthreads();

  if (t < 32) {
    uint32_t kw = (t < 16) ? keepWords[img * 16 + t] : 0u;
    for (int i = 0; i < TOPK; ++i) {
      uint32_t w = __shfl(kw, i >> 5, 32);
      if ((w >> (i & 31)) & 1u) {
        const int wi = i >> 5, bi = i & 31;
        uint32_t gt;
        if (t < wi)       gt = 0u;
        else if (t == wi) gt = (bi == 31) ? 0u : (0xFFFFFFFFu << (bi + 1));
        else              gt = 0xFFFFFFFFu;
        uint32_t mrow = (t < 16) ? rowsL[i * 16 + t] : 0u;
        kw &= ~(mrow & gt);
      }
    }
    if (t < 16) keepL[t] = kw;
  }
  __syncthreads();

  uint32_t bit = (keepL[t >> 5] >> (t & 31)) & 1u;
  size_t o = (size_t)img * TOPK + t;
  float4 bb = boxes[o];
  float cf = confs[o], cl = clsf[o];
  float kf = bit ? 1.0f : 0.0f;
  size_t db = o * 6;
  outDet[db + 0] = kf * bb.x;
  outDet[db + 1] = kf * bb.y;
  outDet[db + 2] = kf * bb.z;
  outDet[db + 3] = kf * bb.w;
  outDet[db + 4] = kf * cf;
  outDet[db + 5] = kf * cl;
  outKeep[o] = kf;
}

// --------------------------------------------------------------------------
static inline size_t alignUp(size_t x) { return (x + 255) & ~(size_t)255; }

extern "C" void kernel_launch(void* const* d_in, const int* in_sizes, int n_in,
                              void* d_out, int out_size, void* d_ws, size_t ws_size,
                              hipStream_t stream) {
  const float* pred = (const float*)d_in[0];

  char* w = (char*)d_ws;
  uint32_t* keys      = (uint32_t*)w;        w += alignUp((size_t)B * N * 4);
  int*      selIdx    = (int*)w;             w += alignUp((size_t)B * TOPK * 4);
  float4*   boxes     = (float4*)w;          w += alignUp((size_t)B * TOPK * 16);
  float4*   nmsB      = (float4*)w;          w += alignUp((size_t)B * TOPK * 16);
  float*    areas     = (float*)w;           w += alignUp((size_t)B * TOPK * 4);
  float*    confs     = (float*)w;           w += alignUp((size_t)B * TOPK * 4);
  float*    clsf      = (float*)w;           w += alignUp((size_t)B * TOPK * 4);
  uint32_t* keepWords = (uint32_t*)w;        w += alignUp((size_t)B * 16 * 4);
  unsigned short* mask = (unsigned short*)w; w += alignUp((size_t)B * TOPK * 32 * 2);

  float* outDet  = (float*)d_out;
  float* outKeep = outDet + (size_t)B * TOPK * 6;

  const int totalWaves = B * N;                  // one wave per anchor
  const int blocks1 = (totalWaves * 32) / 256;   // exact multiple
  k_scores<<<blocks1, 256, 0, stream>>>(pred, keys);
  k_select<<<B, 256, 0, stream>>>(keys, selIdx);
  k_gather<<<B, 512, 0, stream>>>(pred, selIdx, boxes, nmsB, areas, confs, clsf, keepWords);
  k_iou<<<B * 32, 256, 0, stream>>>(nmsB, areas, mask);
  k_nms<<<B, 512, 0, stream>>>(mask, keepWords, boxes, confs, clsf, outDet, outKeep);
}